// ProviderGAT_12781822673352
// MI455X (gfx1250) — compile-verified
//
#include <hip/hip_runtime.h>
#include <math.h>

typedef __attribute__((ext_vector_type(2))) float v2f;
typedef __attribute__((ext_vector_type(8))) float v8f;

#define H       4
#define HID     256
#define CD      512
#define NCODES  200000
#define NEG_SLOPE 0.2f

// workspace layout (float offsets)
#define WS_BFRAG 0        // 128 k-steps * 32 lanes * 2 floats = 8192
#define WS_OFF   8192     // 4 floats: p_score[h] + b2[h]
#define WS_GMAX  8196     // 4
#define WS_GINV  8200     // 4
#define WS_PMAX  8208     // 256 blocks * 4
#define WS_PSUM  9232     // 256 blocks * 4

// ---------------------------------------------------------------------------
// K1: fold the hidden layer.  w2[h][c] = sum_d a2[h][d]*Ws[h][d][c];
// off[h] = p_score[h] + b2[h].  Pre-swizzle w2 into WMMA B-fragment layout.
// ---------------------------------------------------------------------------
__global__ __launch_bounds__(256) void prep_kernel(
    const float* __restrict__ prov,   // (1,512)
    const float* __restrict__ Ws,     // (4,256,512)
    const float* __restrict__ bs,     // (4,256)
    const float* __restrict__ as_,    // (4,512)  [a1 | a2]
    float* __restrict__ ws)
{
  __shared__ float sProv[CD];
  __shared__ float sW2[H][CD];
  __shared__ float sRed[256];
  const int tid = threadIdx.x;

  sProv[tid]       = prov[tid];
  sProv[tid + 256] = prov[tid + 256];
  __syncthreads();

  // w2[h][c]: coalesced over c, strided over d (tiny: 2MB total)
  for (int h = 0; h < H; ++h) {
    const float* a2 = as_ + h * CD + HID;
    for (int c = tid; c < CD; c += 256) {
      const float* wcol = Ws + (size_t)(h * HID) * CD + c;
      float acc = 0.f;
      for (int d = 0; d < HID; ++d) acc += wcol[(size_t)d * CD] * a2[d];
      sW2[h][c] = acc;
    }
  }
  __syncthreads();

  // off[h] = sum_d a1*(Ws[h,d,:].prov + bs) + bs*a2   (thread = d)
  for (int h = 0; h < H; ++h) {
    const int d = tid;
    const float* wrow = Ws + (size_t)(h * HID + d) * CD;
    float t = 0.f;
    for (int c = 0; c < CD; ++c) t += wrow[c] * sProv[c];
    const float b  = bs[h * HID + d];
    const float a1 = as_[h * CD + d];
    const float a2 = as_[h * CD + HID + d];
    sRed[tid] = a1 * (t + b) + a2 * b;
    __syncthreads();
    for (int s = 128; s > 0; s >>= 1) {
      if (tid < s) sRed[tid] += sRed[tid + s];
      __syncthreads();
    }
    if (tid == 0) ws[WS_OFF + h] = sRed[0];
    __syncthreads();
  }

  // B fragments for V_WMMA_F32_16X16X4_F32:
  // lane holds N=lane&15 (head, zero-padded to 16); VGPR0=B[k+off], VGPR1=B[k+off+1],
  // off = 2 for lanes 16..31.  B[k][n] = w2[n][k].
  for (int idx = tid; idx < 4096; idx += 256) {
    const int s = idx >> 5, l = idx & 31;
    const int h = l & 15;
    const int k = 4 * s + ((l >> 4) << 1);
    float x = 0.f, y = 0.f;
    if (h < H) { x = sW2[h][k]; y = sW2[h][k + 1]; }
    ws[WS_BFRAG + idx * 2]     = x;
    ws[WS_BFRAG + idx * 2 + 1] = y;
  }
}

// ---------------------------------------------------------------------------
// K2: the 410MB stream.  e[h][n] = leaky(off[h] + code[n,:].w2[h,:]).
// Block = 256 thr (8 waves) x 128 rows; LDS-staged K-chunks of 64;
// per-wave 16x16 tile via 128x v_wmma_f32_16x16x4_f32.
// ---------------------------------------------------------------------------
__global__ __launch_bounds__(256) void score_kernel(
    const float* __restrict__ code,   // (200000,512)
    const float* __restrict__ ws,
    float* __restrict__ out)          // raw e, (4,200000)
{
  __shared__ float sA[128 * 68];      // 128 rows, stride 68 (16B-aligned, bank-safe)
  const int tid  = threadIdx.x;
  const int wave = tid >> 5;
  const int lane = tid & 31;
  const int m    = lane & 15;         // row-in-tile / head column
  const int half = lane >> 4;         // 0: K+0/1, 1: K+2/3
  const int rowBase = blockIdx.x * 128;

  const float offh = ws[WS_OFF + (m & 3)];
  const float* bfrag = ws + WS_BFRAG;

  v8f acc = {0.f, 0.f, 0.f, 0.f, 0.f, 0.f, 0.f, 0.f};

  for (int k0 = 0; k0 < CD; k0 += 64) {
    // cooperative coalesced load: 128 rows x 64 floats (16 float4 per row)
    #pragma unroll
    for (int i = 0; i < 8; ++i) {
      const int q  = tid + i * 256;   // 0..2047
      const int r  = q >> 4;
      const int c4 = q & 15;
      int gr = rowBase + r;
      if (gr >= NCODES) gr = NCODES - 1;
      const float4 v = *(const float4*)(code + (size_t)gr * CD + k0 + c4 * 4);
      *(float4*)(&sA[r * 68 + c4 * 4]) = v;
    }
    __syncthreads();

    const float* aBase = &sA[((wave << 4) + m) * 68 + (half << 1)];
    #pragma unroll
    for (int kk = 0; kk < 64; kk += 4) {
      v2f a = *(const v2f*)(aBase + kk);                       // ds_load_b64
      const int s = (k0 + kk) >> 2;
      v2f b = *(const v2f*)(bfrag + (size_t)(s * 32 + lane) * 2); // L2-hot 8B load
      acc = __builtin_amdgcn_wmma_f32_16x16x4_f32(
                false, a, false, b, (short)0, acc, false, false);
    }
    __syncthreads();
  }

  // D layout: lane -> col N = m; VGPR v -> row v (+8 for lanes 16..31)
  if (m < H) {
    const int rtop = half << 3;
    #pragma unroll
    for (int v = 0; v < 8; ++v) {
      const int n = rowBase + (wave << 4) + rtop + v;
      if (n < NCODES) {
        float e = offh + acc[v];
        e = (e >= 0.f) ? e : NEG_SLOPE * e;
        out[(size_t)m * NCODES + n] = e;
      }
    }
  }
}

// ---------------------------------------------------------------------------
// K3a/K3b: per-head global max (deterministic fixed-order tree reduction)
// ---------------------------------------------------------------------------
__global__ __launch_bounds__(256) void pmax_kernel(
    const float* __restrict__ out, float* __restrict__ ws)
{
  __shared__ float sm[256 * 4];
  const int tid = threadIdx.x;
  float lm[4] = {-INFINITY, -INFINITY, -INFINITY, -INFINITY};
  for (int idx = blockIdx.x * 256 + tid; idx < H * NCODES; idx += 256 * 256) {
    const int hh = idx / NCODES;
    lm[hh] = fmaxf(lm[hh], out[idx]);
  }
  for (int h = 0; h < 4; ++h) sm[tid * 4 + h] = lm[h];
  __syncthreads();
  for (int s = 128; s > 0; s >>= 1) {
    if (tid < s)
      for (int h = 0; h < 4; ++h)
        sm[tid * 4 + h] = fmaxf(sm[tid * 4 + h], sm[(tid + s) * 4 + h]);
    __syncthreads();
  }
  if (tid < 4) ws[WS_PMAX + blockIdx.x * 4 + tid] = sm[tid];
}

__global__ void rmax_kernel(float* __restrict__ ws)
{
  const int tid = threadIdx.x;
  if (tid < 4) {
    float mx = -INFINITY;
    for (int i = 0; i < 256; ++i) mx = fmaxf(mx, ws[WS_PMAX + i * 4 + tid]);
    ws[WS_GMAX + tid] = mx;
  }
}

// ---------------------------------------------------------------------------
// K4a/K4b: exp(e - max) in place + per-head sum; then invert sums
// ---------------------------------------------------------------------------
__global__ __launch_bounds__(256) void psum_kernel(
    float* __restrict__ out, float* __restrict__ ws)
{
  __shared__ float sm[256 * 4];
  __shared__ float gmax[4];
  const int tid = threadIdx.x;
  if (tid < 4) gmax[tid] = ws[WS_GMAX + tid];
  __syncthreads();
  float lsum[4] = {0.f, 0.f, 0.f, 0.f};
  for (int idx = blockIdx.x * 256 + tid; idx < H * NCODES; idx += 256 * 256) {
    const int hh = idx / NCODES;
    const float ex = expf(out[idx] - gmax[hh]);
    out[idx] = ex;
    lsum[hh] += ex;
  }
  for (int h = 0; h < 4; ++h) sm[tid * 4 + h] = lsum[h];
  __syncthreads();
  for (int s = 128; s > 0; s >>= 1) {
    if (tid < s)
      for (int h = 0; h < 4; ++h)
        sm[tid * 4 + h] += sm[(tid + s) * 4 + h];
    __syncthreads();
  }
  if (tid < 4) ws[WS_PSUM + blockIdx.x * 4 + tid] = sm[tid];
}

__global__ void rsum_kernel(float* __restrict__ ws)
{
  const int tid = threadIdx.x;
  if (tid < 4) {
    float s = 0.f;
    for (int i = 0; i < 256; ++i) s += ws[WS_PSUM + i * 4 + tid];
    ws[WS_GINV + tid] = 1.0f / s;
  }
}

// ---------------------------------------------------------------------------
// K5: normalize in place
// ---------------------------------------------------------------------------
__global__ __launch_bounds__(256) void norm_kernel(
    float* __restrict__ out, const float* __restrict__ ws)
{
  const int idx = blockIdx.x * 256 + threadIdx.x;
  if (idx < H * NCODES) out[idx] *= ws[WS_GINV + idx / NCODES];
}

// ---------------------------------------------------------------------------
extern "C" void kernel_launch(void* const* d_in, const int* in_sizes, int n_in,
                              void* d_out, int out_size, void* d_ws, size_t ws_size,
                              hipStream_t stream)
{
  const float* prov = (const float*)d_in[0];   // provider_emb (1,512)
  const float* code = (const float*)d_in[1];   // code_embs (200000,512)
  const float* Ws   = (const float*)d_in[2];   // (4,256,512)
  const float* bs   = (const float*)d_in[3];   // (4,256)
  const float* as_  = (const float*)d_in[4];   // (4,512)
  float* out = (float*)d_out;                  // alpha (4,200000)
  float* ws  = (float*)d_ws;

  prep_kernel<<<1, 256, 0, stream>>>(prov, Ws, bs, as_, ws);

  const int nTileBlocks = (NCODES + 127) / 128;   // 1563
  score_kernel<<<nTileBlocks, 256, 0, stream>>>(code, ws, out);

  pmax_kernel<<<256, 256, 0, stream>>>(out, ws);
  rmax_kernel<<<1, 64, 0, stream>>>(ws);
  psum_kernel<<<256, 256, 0, stream>>>(out, ws);
  rsum_kernel<<<1, 64, 0, stream>>>(ws);
  norm_kernel<<<(H * NCODES + 255) / 256, 256, 0, stream>>>(out, ws);
}